// ABGNN_11708080849339
// MI455X (gfx1250) — compile-verified
//
#include <hip/hip_runtime.h>
#include <stdint.h>

typedef __attribute__((ext_vector_type(2))) float v2f;
typedef __attribute__((ext_vector_type(8))) float v8f;

#define HID 128
#define IN_DIM 16
#define DEG 16

// ---------------------------------------------------------------------------
// CDNA5 async DMA: global -> LDS, 16 bytes per lane, tracked by ASYNCcnt.
// lds_off is the wave-relative LDS byte address (low 32 bits of a generic
// pointer into __shared__), gaddr a 64-bit global address.
// ---------------------------------------------------------------------------
__device__ __forceinline__ void async_copy_b128(uint32_t lds_off,
                                                const float* gaddr) {
  asm volatile("global_load_async_to_lds_b128 %0, %1, off"
               :
               : "v"(lds_off), "v"(gaddr)
               : "memory");
}

__device__ __forceinline__ void wait_async_all() {
  asm volatile("s_wait_asynccnt 0x0" ::: "memory");
}

// ---------------------------------------------------------------------------
// h0 = relu(features @ W_init + b_init)
// One 16-row tile per block; 8 waves each own a 16-column tile. K=16 -> 4 WMMA.
// A-frag (32-bit, 16x4): lane l holds M=l&15, K pair = (l>>4)*2 + {0,1}.
// B-frag (4x16): lane l holds N=l&15, same K pair striping.
// ---------------------------------------------------------------------------
__global__ __launch_bounds__(256) void k_init_gemm(
    const float* __restrict__ feat, const float* __restrict__ W,
    const float* __restrict__ bias, float* __restrict__ h0) {
  const int row0 = blockIdx.x * 16;
  const int lane = threadIdx.x & 31;
  const int wave = threadIdx.x >> 5;      // 0..7 -> column tile
  const int col0 = wave * 16;
  const int mn   = lane & 15;             // M for A, N for B/D
  const int kh   = lane >> 4;             // which K half-pair this lane holds

  v8f c = {};
#pragma unroll
  for (int kb = 0; kb < IN_DIM; kb += 4) {
    const int k0 = kb + kh * 2;
    v2f a = *(const v2f*)(feat + (size_t)(row0 + mn) * IN_DIM + k0);
    v2f b;
    b.x = W[(size_t)(k0    ) * HID + col0 + mn];
    b.y = W[(size_t)(k0 + 1) * HID + col0 + mn];
    c = __builtin_amdgcn_wmma_f32_16x16x4_f32(false, a, false, b, (short)0, c,
                                              false, false);
  }
  const float bv = bias[col0 + mn];
#pragma unroll
  for (int r = 0; r < 8; ++r) {
    const int row = row0 + r + 8 * kh;    // D layout: VGPR r, lane-half kh
    float v = c[r] + bv;
    h0[(size_t)row * HID + col0 + mn] = v > 0.f ? v : 0.f;
  }
}

// ---------------------------------------------------------------------------
// neigh[i,:] = mean over edges e with dst[e]==i of h[src[e],:]
// dst is sorted -> per-node range found by binary search (deterministic,
// no atomics). 128 threads = 128 features, fully coalesced row gathers that
// hit the 192MB L2 (h0 table is 102MB).
// ---------------------------------------------------------------------------
__global__ __launch_bounds__(128) void k_seg_mean(
    const float* __restrict__ h, const int* __restrict__ src,
    const int* __restrict__ dst, int nEdges, float* __restrict__ neigh) {
  const int node = blockIdx.x;
  const int f    = threadIdx.x;

  int lo = 0, hi = nEdges;                      // lower_bound(node)
  while (lo < hi) { int mid = (lo + hi) >> 1; if (dst[mid] < node) lo = mid + 1; else hi = mid; }
  const int beg = lo;
  hi = nEdges;                                  // upper_bound(node)
  while (lo < hi) { int mid = (lo + hi) >> 1; if (dst[mid] <= node) lo = mid + 1; else hi = mid; }
  const int end = lo;

  float acc = 0.f;
  for (int e = beg; e < end; ++e)
    acc += h[(size_t)src[e] * HID + f];

  const float cnt = (float)(end - beg);
  neigh[(size_t)node * HID + f] = acc / fmaxf(cnt, 1.0f);
}

// ---------------------------------------------------------------------------
// out = act( hprev[:n] @ W_self + b_self + neigh @ W_neigh + b_neigh )
// A-tiles (16x128 each) staged in LDS with GLOBAL_LOAD_ASYNC_TO_LDS_B128
// (ASYNCcnt-tracked DMA, no VGPR round-trip); both GEMMs accumulate into the
// same f32 WMMA accumulator: 64 v_wmma per wave.
// ---------------------------------------------------------------------------
__global__ __launch_bounds__(256) void k_combine(
    const float* __restrict__ hprev, const float* __restrict__ neigh,
    const float* __restrict__ Wself, const float* __restrict__ bself,
    const float* __restrict__ Wneigh, const float* __restrict__ bneigh,
    float* __restrict__ out, int act) {
  __shared__ float As[16 * HID];
  __shared__ float An[16 * HID];

  const int row0 = blockIdx.x * 16;
  const int tid  = threadIdx.x;

  // Async stage: 2 tiles x 8KB = 2 x 512 x b128; 256 threads issue 2 each.
  {
    const float*   gs  = hprev + (size_t)row0 * HID;
    const float*   gn  = neigh + (size_t)row0 * HID;
    const uint32_t lAs = (uint32_t)(uintptr_t)As;   // LDS byte offset
    const uint32_t lAn = (uint32_t)(uintptr_t)An;
    async_copy_b128(lAs + (uint32_t)tid * 16u,         gs + (size_t)tid * 4);
    async_copy_b128(lAs + (uint32_t)(tid + 256) * 16u, gs + (size_t)(tid + 256) * 4);
    async_copy_b128(lAn + (uint32_t)tid * 16u,         gn + (size_t)tid * 4);
    async_copy_b128(lAn + (uint32_t)(tid + 256) * 16u, gn + (size_t)(tid + 256) * 4);
  }

  const int lane = tid & 31;
  const int wave = tid >> 5;
  const int col0 = wave * 16;
  const int mn   = lane & 15;
  const int kh   = lane >> 4;

  // Warm the B-operand streams (L2/WGP$-resident weights).
  __builtin_prefetch(Wself + col0, 0, 3);
  __builtin_prefetch(Wneigh + col0, 0, 3);

  wait_async_all();     // own wave's DMAs landed in LDS
  __syncthreads();      // all waves' DMAs visible block-wide

  v8f c = {};
#pragma unroll 4
  for (int kb = 0; kb < HID; kb += 4) {
    const int k0 = kb + kh * 2;
    v2f as = *(const v2f*)(As + mn * HID + k0);
    v2f an = *(const v2f*)(An + mn * HID + k0);
    v2f bs, bn;
    bs.x = Wself [(size_t)(k0    ) * HID + col0 + mn];
    bs.y = Wself [(size_t)(k0 + 1) * HID + col0 + mn];
    bn.x = Wneigh[(size_t)(k0    ) * HID + col0 + mn];
    bn.y = Wneigh[(size_t)(k0 + 1) * HID + col0 + mn];
    c = __builtin_amdgcn_wmma_f32_16x16x4_f32(false, as, false, bs, (short)0, c,
                                              false, false);
    c = __builtin_amdgcn_wmma_f32_16x16x4_f32(false, an, false, bn, (short)0, c,
                                              false, false);
  }

  const float bv = bself[col0 + mn] + bneigh[col0 + mn];
#pragma unroll
  for (int r = 0; r < 8; ++r) {
    const int row = row0 + r + 8 * kh;
    float v = c[r] + bv;
    if (act) v = v > 0.f ? v : 0.f;
    out[(size_t)row * HID + col0 + mn] = v;
  }
}

// ---------------------------------------------------------------------------
extern "C" void kernel_launch(void* const* d_in, const int* in_sizes, int n_in,
                              void* d_out, int out_size, void* d_ws, size_t ws_size,
                              hipStream_t stream) {
  const float* features = (const float*)d_in[0];
  const int*   src0     = (const int*)  d_in[1];
  const int*   dst0     = (const int*)  d_in[2];
  const int*   src1     = (const int*)  d_in[3];
  const int*   dst1     = (const int*)  d_in[4];
  const float* W_init   = (const float*)d_in[5];
  const float* b_init   = (const float*)d_in[6];
  const float* W_self   = (const float*)d_in[7];
  const float* b_self   = (const float*)d_in[8];
  const float* W_neigh  = (const float*)d_in[9];
  const float* b_neigh  = (const float*)d_in[10];

  const int N0 = in_sizes[0] / IN_DIM;   // 200000
  const int E1 = in_sizes[1];            // N1*DEG
  const int E2 = in_sizes[3];            // N2*DEG
  const int N1 = E1 / DEG;               // 100000
  const int N2 = E2 / DEG;               // 50000

  float* h0 = (float*)d_ws;                      // [N0,128]
  float* h1 = h0 + (size_t)N0 * HID;             // [N1,128]
  float* ng = h1 + (size_t)N1 * HID;             // [N1,128] (reused for level 2)

  // Layer 0: h0 = relu(features @ W_init + b_init)
  k_init_gemm<<<N0 / 16, 256, 0, stream>>>(features, W_init, b_init, h0);

  // Block 1 (relu)
  k_seg_mean<<<N1, 128, 0, stream>>>(h0, src0, dst0, E1, ng);
  k_combine<<<N1 / 16, 256, 0, stream>>>(h0, ng, W_self, b_self, W_neigh,
                                         b_neigh, h1, 1);

  // Block 2 (no activation) -> d_out
  k_seg_mean<<<N2, 128, 0, stream>>>(h1, src1, dst1, E2, ng);
  k_combine<<<N2 / 16, 256, 0, stream>>>(h1, ng, W_self, b_self, W_neigh,
                                         b_neigh, (float*)d_out, 0);
}